// STGAT_discriminator_64604898067035
// MI455X (gfx1250) — compile-verified
//
#include <hip/hip_runtime.h>
#include <math.h>

typedef float v2f __attribute__((ext_vector_type(2)));
typedef float v8f __attribute__((ext_vector_type(8)));

#define TT 20
#define SS 64
#define PP 64
#define NN (SS*PP)
#define EPSV 1e-5f

__device__ __forceinline__ v8f wmma_f32(v2f a, v2f b, v8f c) {
    // D = A(16x4) * B(4x16) + C(16x16), all f32, wave32
    return __builtin_amdgcn_wmma_f32_16x16x4_f32(false, a, false, b, (short)0, c, false, false);
}

__device__ __forceinline__ float sigmoidf_(float x) { return 1.0f / (1.0f + expf(-x)); }

// ---------------------------------------------------------------------------
// Kernel 1: traj LSTM. 1 wave per 16 rows; all 20 steps in-kernel.
// gates[16,128] = x[16,2]@Wih.T + h[16,32]@Whh.T + b  (recurrent part via WMMA)
// ---------------------------------------------------------------------------
__global__ void __launch_bounds__(32)
traj_lstm_kernel(const float* __restrict__ obs,   // [N, 2T]
                 const float* __restrict__ h0, const float* __restrict__ c0,
                 const float* __restrict__ Wih,   // [128,2]
                 const float* __restrict__ Whh,   // [128,32]
                 const float* __restrict__ bih, const float* __restrict__ bhh,
                 float* __restrict__ hs)          // [T,N,32]
{
    __shared__ float h_lds[16 * 32];
    __shared__ float x_lds[16 * 2];
    const int lane = threadIdx.x;
    const int col = lane & 15, hi = lane >> 4;
    const int r0 = blockIdx.x * 16;

    for (int i = lane; i < 16 * 32; i += 32) h_lds[i] = h0[r0 * 32 + i];
    v8f c_reg[2];
    #pragma unroll
    for (int j = 0; j < 2; ++j)
        #pragma unroll
        for (int v = 0; v < 8; ++v)
            c_reg[j][v] = c0[(r0 + v + 8 * hi) * 32 + j * 16 + col];
    __syncthreads();

    for (int t = 0; t < TT; ++t) {
        x_lds[lane] = obs[(r0 + (lane >> 1)) * (2 * TT) + t * 2 + (lane & 1)];
        __syncthreads();

        v8f acc[8];
        #pragma unroll
        for (int nt = 0; nt < 8; ++nt) {
            const int n = nt * 16 + col;
            const float binit = bih[n] + bhh[n];
            const float wx0 = Wih[n * 2 + 0], wx1 = Wih[n * 2 + 1];
            #pragma unroll
            for (int v = 0; v < 8; ++v) {
                const int m = v + 8 * hi;
                acc[nt][v] = binit + x_lds[m * 2] * wx0 + x_lds[m * 2 + 1] * wx1;
            }
            #pragma unroll
            for (int kc = 0; kc < 8; ++kc) {
                const int kb = kc * 4 + 2 * hi;
                v2f a, b;
                a.x = h_lds[col * 32 + kb]; a.y = h_lds[col * 32 + kb + 1];
                b.x = Whh[n * 32 + kb];     b.y = Whh[n * 32 + kb + 1];
                acc[nt] = wmma_f32(a, b, acc[nt]);
            }
        }
        __syncthreads();
        #pragma unroll
        for (int j = 0; j < 2; ++j)
            #pragma unroll
            for (int v = 0; v < 8; ++v) {
                const float ig = acc[j][v], fg = acc[2 + j][v];
                const float gg = acc[4 + j][v], og = acc[6 + j][v];
                const float cc = sigmoidf_(fg) * c_reg[j][v] + sigmoidf_(ig) * tanhf(gg);
                c_reg[j][v] = cc;
                const float hh = sigmoidf_(og) * tanhf(cc);
                const int m = v + 8 * hi;
                h_lds[m * 32 + j * 16 + col] = hh;
                hs[(size_t)t * NN * 32 + (size_t)(r0 + m) * 32 + j * 16 + col] = hh;
            }
        __syncthreads();
    }
}

// ---------------------------------------------------------------------------
// Kernel 2: GAT encoder. One block (4 waves) per (t, scene); P=64.
// inst_norm -> GAT0 (4 heads) -> ELU -> inst_norm -> GAT1 -> graph_in
// ---------------------------------------------------------------------------
__global__ void __launch_bounds__(128)
gat_kernel(const float* __restrict__ hs,   // [T,N,32]
           const float* __restrict__ w0,   // [4,32,16]
           const float* __restrict__ as0, const float* __restrict__ ad0,  // [4,16]
           const float* __restrict__ b0,   // [16]
           const float* __restrict__ w1,   // [64,32]
           const float* __restrict__ as1, const float* __restrict__ ad1,  // [32]
           const float* __restrict__ b1,   // [32]
           float* __restrict__ gout)       // [T,N,32]
{
    __shared__ float xn[64 * 32];   // normalized layer-1 input
    __shared__ float hp[64 * 64];   // projections (layer2 uses stride 32)
    __shared__ float yb[64 * 64];   // layer-1 output / layer-2 input
    __shared__ float at[64 * 64];   // attention (one head at a time)
    __shared__ float es[4 * 64];
    __shared__ float ed[4 * 64];

    const int tid = threadIdx.x;
    const int lane = tid & 31, w = tid >> 5;
    const int col = lane & 15, hi = lane >> 4;
    const int t = blockIdx.x / SS, s = blockIdx.x % SS;
    const size_t base = (size_t)t * NN * 32 + (size_t)s * PP * 32;
    const int m0 = w * 16;

    for (int i = tid; i < 64 * 32; i += 128) xn[i] = hs[base + i];
    __syncthreads();

    // instance norm over pedestrians, per channel (32 ch)
    if (tid < 32) {
        float m = 0.f, m2 = 0.f;
        for (int p = 0; p < 64; ++p) { float v = xn[p * 32 + tid]; m += v; m2 += v * v; }
        m *= (1.f / 64.f); m2 *= (1.f / 64.f);
        const float inv = rsqrtf(m2 - m * m + EPSV);
        for (int p = 0; p < 64; ++p) xn[p * 32 + tid] = (xn[p * 32 + tid] - m) * inv;
    }
    __syncthreads();

    // hp[64,64] = xn[64,32] @ W0concat[32,64]  (col = head*16 + o)
    #pragma unroll
    for (int nt = 0; nt < 4; ++nt) {
        const int n = nt * 16 + col;
        const int h = n >> 4, o = n & 15;
        v8f acc = {};
        #pragma unroll
        for (int kc = 0; kc < 8; ++kc) {
            const int kb = kc * 4 + 2 * hi;
            v2f a, b;
            a.x = xn[(m0 + col) * 32 + kb]; a.y = xn[(m0 + col) * 32 + kb + 1];
            b.x = w0[h * 512 + kb * 16 + o]; b.y = w0[h * 512 + (kb + 1) * 16 + o];
            acc = wmma_f32(a, b, acc);
        }
        #pragma unroll
        for (int v = 0; v < 8; ++v) hp[(m0 + v + 8 * hi) * 64 + n] = acc[v];
    }
    __syncthreads();

    // attention logits per head
    for (int e = tid; e < 256; e += 128) {
        const int h = e >> 6, p = e & 63;
        float s1 = 0.f, s2 = 0.f;
        for (int o = 0; o < 16; ++o) {
            const float v = hp[p * 64 + h * 16 + o];
            s1 += v * as0[h * 16 + o];
            s2 += v * ad0[h * 16 + o];
        }
        es[e] = s1; ed[e] = s2;
    }
    __syncthreads();

    for (int h = 0; h < 4; ++h) {
        if (tid < 64) {
            const int p = tid;
            const float esp = es[h * 64 + p];
            float mx = -1e30f;
            for (int q = 0; q < 64; ++q) {
                float z = esp + ed[h * 64 + q];
                z = z > 0.f ? z : 0.2f * z;
                mx = fmaxf(mx, z);
            }
            float sum = 0.f;
            for (int q = 0; q < 64; ++q) {
                float z = esp + ed[h * 64 + q];
                z = z > 0.f ? z : 0.2f * z;
                const float e = expf(z - mx);
                at[p * 64 + q] = e;
                sum += e;
            }
            const float inv = 1.f / sum;
            for (int q = 0; q < 64; ++q) at[p * 64 + q] *= inv;
        }
        __syncthreads();

        // out[:, h*16: h*16+16] = attn @ hp[:, h*16:...]  + b0, then ELU
        v8f acc;
        #pragma unroll
        for (int v = 0; v < 8; ++v) acc[v] = b0[col];
        #pragma unroll
        for (int kc = 0; kc < 16; ++kc) {
            const int kb = kc * 4 + 2 * hi;
            v2f a, b;
            a.x = at[(m0 + col) * 64 + kb]; a.y = at[(m0 + col) * 64 + kb + 1];
            b.x = hp[kb * 64 + h * 16 + col]; b.y = hp[(kb + 1) * 64 + h * 16 + col];
            acc = wmma_f32(a, b, acc);
        }
        #pragma unroll
        for (int v = 0; v < 8; ++v) {
            float z = acc[v];
            z = z > 0.f ? z : (expf(z) - 1.f);   // ELU
            yb[(m0 + v + 8 * hi) * 64 + h * 16 + col] = z;
        }
        __syncthreads();
    }

    // instance norm (64 ch)
    if (tid < 64) {
        float m = 0.f, m2 = 0.f;
        for (int p = 0; p < 64; ++p) { float v = yb[p * 64 + tid]; m += v; m2 += v * v; }
        m *= (1.f / 64.f); m2 *= (1.f / 64.f);
        const float inv = rsqrtf(m2 - m * m + EPSV);
        for (int p = 0; p < 64; ++p) yb[p * 64 + tid] = (yb[p * 64 + tid] - m) * inv;
    }
    __syncthreads();

    // hp1[64,32] = yb[64,64] @ w1[64,32]  (stored in hp with stride 32)
    #pragma unroll
    for (int nt = 0; nt < 2; ++nt) {
        const int n = nt * 16 + col;
        v8f acc = {};
        #pragma unroll
        for (int kc = 0; kc < 16; ++kc) {
            const int kb = kc * 4 + 2 * hi;
            v2f a, b;
            a.x = yb[(m0 + col) * 64 + kb]; a.y = yb[(m0 + col) * 64 + kb + 1];
            b.x = w1[kb * 32 + n];          b.y = w1[(kb + 1) * 32 + n];
            acc = wmma_f32(a, b, acc);
        }
        #pragma unroll
        for (int v = 0; v < 8; ++v) hp[(m0 + v + 8 * hi) * 32 + n] = acc[v];
    }
    __syncthreads();

    if (tid < 64) {
        const int p = tid;
        float s1 = 0.f, s2 = 0.f;
        for (int o = 0; o < 32; ++o) {
            const float v = hp[p * 32 + o];
            s1 += v * as1[o]; s2 += v * ad1[o];
        }
        es[p] = s1; ed[p] = s2;
    }
    __syncthreads();
    if (tid < 64) {
        const int p = tid;
        const float esp = es[p];
        float mx = -1e30f;
        for (int q = 0; q < 64; ++q) {
            float z = esp + ed[q];
            z = z > 0.f ? z : 0.2f * z;
            mx = fmaxf(mx, z);
        }
        float sum = 0.f;
        for (int q = 0; q < 64; ++q) {
            float z = esp + ed[q];
            z = z > 0.f ? z : 0.2f * z;
            const float e = expf(z - mx);
            at[p * 64 + q] = e;
            sum += e;
        }
        const float inv = 1.f / sum;
        for (int q = 0; q < 64; ++q) at[p * 64 + q] *= inv;
    }
    __syncthreads();

    // out1[64,32] = attn1 @ hp1 + b1 -> graph_in
    #pragma unroll
    for (int nt = 0; nt < 2; ++nt) {
        const int n = nt * 16 + col;
        v8f acc;
        #pragma unroll
        for (int v = 0; v < 8; ++v) acc[v] = b1[n];
        #pragma unroll
        for (int kc = 0; kc < 16; ++kc) {
            const int kb = kc * 4 + 2 * hi;
            v2f a, b;
            a.x = at[(m0 + col) * 64 + kb]; a.y = at[(m0 + col) * 64 + kb + 1];
            b.x = hp[kb * 32 + n];          b.y = hp[(kb + 1) * 32 + n];
            acc = wmma_f32(a, b, acc);
        }
        #pragma unroll
        for (int v = 0; v < 8; ++v)
            gout[base + (size_t)(m0 + v + 8 * hi) * 32 + n] = acc[v];
    }
}

// ---------------------------------------------------------------------------
// Kernel 3: graph LSTM (input 32, hidden 32). K=64 fused (input + recurrent).
// ---------------------------------------------------------------------------
__global__ void __launch_bounds__(32)
graph_lstm_kernel(const float* __restrict__ gin,  // [T,N,32]
                  const float* __restrict__ h0, const float* __restrict__ c0,
                  const float* __restrict__ Wih,  // [128,32]
                  const float* __restrict__ Whh,  // [128,32]
                  const float* __restrict__ bih, const float* __restrict__ bhh,
                  float* __restrict__ hT)         // [N,32]
{
    __shared__ float h_lds[16 * 32];
    const int lane = threadIdx.x;
    const int col = lane & 15, hi = lane >> 4;
    const int r0 = blockIdx.x * 16;

    for (int i = lane; i < 16 * 32; i += 32) h_lds[i] = h0[r0 * 32 + i];
    v8f c_reg[2];
    #pragma unroll
    for (int j = 0; j < 2; ++j)
        #pragma unroll
        for (int v = 0; v < 8; ++v)
            c_reg[j][v] = c0[(r0 + v + 8 * hi) * 32 + j * 16 + col];
    __syncthreads();

    for (int t = 0; t < TT; ++t) {
        const float* xt = gin + (size_t)t * NN * 32;
        v8f acc[8];
        #pragma unroll
        for (int nt = 0; nt < 8; ++nt) {
            const int n = nt * 16 + col;
            const float binit = bih[n] + bhh[n];
            #pragma unroll
            for (int v = 0; v < 8; ++v) acc[nt][v] = binit;
            #pragma unroll
            for (int kc = 0; kc < 8; ++kc) {   // input contribution
                const int kb = kc * 4 + 2 * hi;
                v2f a, b;
                a.x = xt[(r0 + col) * 32 + kb]; a.y = xt[(r0 + col) * 32 + kb + 1];
                b.x = Wih[n * 32 + kb];         b.y = Wih[n * 32 + kb + 1];
                acc[nt] = wmma_f32(a, b, acc[nt]);
            }
            #pragma unroll
            for (int kc = 0; kc < 8; ++kc) {   // recurrent contribution
                const int kb = kc * 4 + 2 * hi;
                v2f a, b;
                a.x = h_lds[col * 32 + kb]; a.y = h_lds[col * 32 + kb + 1];
                b.x = Whh[n * 32 + kb];     b.y = Whh[n * 32 + kb + 1];
                acc[nt] = wmma_f32(a, b, acc[nt]);
            }
        }
        __syncthreads();
        #pragma unroll
        for (int j = 0; j < 2; ++j)
            #pragma unroll
            for (int v = 0; v < 8; ++v) {
                const float ig = acc[j][v], fg = acc[2 + j][v];
                const float gg = acc[4 + j][v], og = acc[6 + j][v];
                const float cc = sigmoidf_(fg) * c_reg[j][v] + sigmoidf_(ig) * tanhf(gg);
                c_reg[j][v] = cc;
                h_lds[(v + 8 * hi) * 32 + j * 16 + col] = sigmoidf_(og) * tanhf(cc);
            }
        __syncthreads();
    }
    for (int i = lane; i < 16 * 32; i += 32) hT[r0 * 32 + i] = h_lds[i];
}

// ---------------------------------------------------------------------------
// Kernel 4: pred LSTM (input 2, hidden 80) + output projection.
// ---------------------------------------------------------------------------
__global__ void __launch_bounds__(32)
pred_lstm_kernel(const float* __restrict__ obs,      // [N,2T]
                 const float* __restrict__ trajhT,   // [N,32]
                 const float* __restrict__ graphhT,  // [N,32]
                 const float* __restrict__ noise,    // [S,16]
                 const float* __restrict__ Wih,      // [320,2]
                 const float* __restrict__ Whh,      // [320,80]
                 const float* __restrict__ bih, const float* __restrict__ bhh,
                 const float* __restrict__ Wp,       // [80]
                 const float* __restrict__ bp,       // [1]
                 float* __restrict__ am)             // [T*N]
{
    __shared__ float h_lds[16 * 80];
    __shared__ float x_lds[16 * 2];
    const int lane = threadIdx.x;
    const int col = lane & 15, hi = lane >> 4;
    const int r0 = blockIdx.x * 16;

    for (int i = lane; i < 16 * 80; i += 32) {
        const int m = i / 80, k = i % 80;
        const int row = r0 + m;
        float v;
        if (k < 32)       v = trajhT[row * 32 + k];
        else if (k < 64)  v = graphhT[row * 32 + (k - 32)];
        else              v = noise[(row >> 6) * 16 + (k - 64)];
        h_lds[i] = v;
    }
    v8f c_reg[5];
    #pragma unroll
    for (int j = 0; j < 5; ++j)
        #pragma unroll
        for (int v = 0; v < 8; ++v) c_reg[j][v] = 0.f;
    __syncthreads();

    for (int t = 0; t < TT; ++t) {
        x_lds[lane] = obs[(r0 + (lane >> 1)) * (2 * TT) + t * 2 + (lane & 1)];
        __syncthreads();

        v8f h_new[5];
        #pragma unroll
        for (int j = 0; j < 5; ++j) {
            v8f gacc[4];
            const int ntb[4] = { j, j + 5, j + 10, j + 15 };  // i,f,g,o tiles
            #pragma unroll
            for (int g = 0; g < 4; ++g) {
                const int n = ntb[g] * 16 + col;
                const float binit = bih[n] + bhh[n];
                const float wx0 = Wih[n * 2], wx1 = Wih[n * 2 + 1];
                v8f acc;
                #pragma unroll
                for (int v = 0; v < 8; ++v) {
                    const int m = v + 8 * hi;
                    acc[v] = binit + x_lds[m * 2] * wx0 + x_lds[m * 2 + 1] * wx1;
                }
                #pragma unroll
                for (int kc = 0; kc < 20; ++kc) {
                    const int kb = kc * 4 + 2 * hi;
                    v2f a, b;
                    a.x = h_lds[col * 80 + kb]; a.y = h_lds[col * 80 + kb + 1];
                    b.x = Whh[n * 80 + kb];     b.y = Whh[n * 80 + kb + 1];
                    acc = wmma_f32(a, b, acc);
                }
                gacc[g] = acc;
            }
            #pragma unroll
            for (int v = 0; v < 8; ++v) {
                const float cc = sigmoidf_(gacc[1][v]) * c_reg[j][v]
                               + sigmoidf_(gacc[0][v]) * tanhf(gacc[2][v]);
                c_reg[j][v] = cc;
                h_new[j][v] = sigmoidf_(gacc[3][v]) * tanhf(cc);
            }
        }
        __syncthreads();   // all reads of old h done
        #pragma unroll
        for (int j = 0; j < 5; ++j)
            #pragma unroll
            for (int v = 0; v < 8; ++v)
                h_lds[(v + 8 * hi) * 80 + j * 16 + col] = h_new[j][v];
        __syncthreads();

        if (lane < 16) {
            float sum = bp[0];
            for (int k = 0; k < 80; ++k) sum += h_lds[lane * 80 + k] * Wp[k];
            am[(size_t)t * NN + r0 + lane] = sum;
        }
        __syncthreads();
    }
}

__global__ void std_fill_kernel(const float* __restrict__ als, float* __restrict__ out) {
    const int i = blockIdx.x * blockDim.x + threadIdx.x;
    if (i < TT * NN) {
        const float ls = als[0];
        out[(size_t)TT * NN + i] = ls;
        out[(size_t)2 * TT * NN + i] = expf(ls);
    }
}

extern "C" void kernel_launch(void* const* d_in, const int* in_sizes, int n_in,
                              void* d_out, int out_size, void* d_ws, size_t ws_size,
                              hipStream_t stream) {
    const float* obs   = (const float*)d_in[0];
    const float* th0   = (const float*)d_in[2];
    const float* tc0   = (const float*)d_in[3];
    const float* gh0   = (const float*)d_in[4];
    const float* gc0   = (const float*)d_in[5];
    const float* noise = (const float*)d_in[6];
    const float* w0    = (const float*)d_in[7];
    const float* as0   = (const float*)d_in[8];
    const float* ad0   = (const float*)d_in[9];
    const float* b0    = (const float*)d_in[10];
    const float* w1    = (const float*)d_in[11];
    const float* as1   = (const float*)d_in[12];
    const float* ad1   = (const float*)d_in[13];
    const float* b1    = (const float*)d_in[14];
    const float* Wih_t = (const float*)d_in[15];
    const float* Whh_t = (const float*)d_in[16];
    const float* bih_t = (const float*)d_in[17];
    const float* bhh_t = (const float*)d_in[18];
    const float* Wih_g = (const float*)d_in[19];
    const float* Whh_g = (const float*)d_in[20];
    const float* bih_g = (const float*)d_in[21];
    const float* bhh_g = (const float*)d_in[22];
    const float* Wih_p = (const float*)d_in[23];
    const float* Whh_p = (const float*)d_in[24];
    const float* bih_p = (const float*)d_in[25];
    const float* bhh_p = (const float*)d_in[26];
    const float* Wp    = (const float*)d_in[27];
    const float* bp    = (const float*)d_in[28];
    const float* als   = (const float*)d_in[29];

    float* ws  = (float*)d_ws;
    float* hs  = ws;                                // [T,N,32]
    float* gin = hs + (size_t)TT * NN * 32;         // [T,N,32]
    float* ghT = gin + (size_t)TT * NN * 32;        // [N,32]
    float* out = (float*)d_out;                     // [am | log_std | std]

    traj_lstm_kernel<<<NN / 16, 32, 0, stream>>>(obs, th0, tc0, Wih_t, Whh_t, bih_t, bhh_t, hs);
    gat_kernel<<<TT * SS, 128, 0, stream>>>(hs, w0, as0, ad0, b0, w1, as1, ad1, b1, gin);
    graph_lstm_kernel<<<NN / 16, 32, 0, stream>>>(gin, gh0, gc0, Wih_g, Whh_g, bih_g, bhh_g, ghT);
    pred_lstm_kernel<<<NN / 16, 32, 0, stream>>>(obs, hs + (size_t)(TT - 1) * NN * 32, ghT, noise,
                                                 Wih_p, Whh_p, bih_p, bhh_p, Wp, bp, out);
    std_fill_kernel<<<(TT * NN + 255) / 256, 256, 0, stream>>>(als, out);
}